// EfficientDet_17188459118937
// MI455X (gfx1250) — compile-verified
//
#include <hip/hip_runtime.h>
#include <hip/hip_bf16.h>
#include <stdint.h>

#define BATCH   16
#define NCLS    90
#define NROWS   (BATCH*NCLS)
#define CAP     2048
#define NBINS   128
#define TOPK    200
#define TOPKP   208
#define NTILE   13
#define MAXDET  100
#define CHUNK   65536
#define LOGIT_THR (-2.9444389791664403f)

typedef float        v2f   __attribute__((ext_vector_type(2)));
typedef float        v8f   __attribute__((ext_vector_type(8)));
typedef unsigned int u32x4 __attribute__((ext_vector_type(4)));
typedef int          i32x8 __attribute__((ext_vector_type(8)));
typedef int          i32x4 __attribute__((ext_vector_type(4)));

#if defined(__HIP_DEVICE_COMPILE__) && __has_builtin(__builtin_amdgcn_tensor_load_to_lds)
#define HAVE_TDM 1
#else
#define HAVE_TDM 0
#endif
#if defined(__HIP_DEVICE_COMPILE__) && __has_builtin(__builtin_amdgcn_wmma_f32_16x16x4_f32)
#define HAVE_WMMA4 1
#else
#define HAVE_WMMA4 0
#endif

// Monotone mapping float -> uint (order preserving), bin = key >> 25 (128 bins).
__device__ __forceinline__ unsigned int monokey(float f) {
  unsigned int u = __float_as_uint(f);
  return (u & 0x80000000u) ? ~u : (u | 0x80000000u);
}

// Fast reciprocal: single v_rcp_f32 instead of the IEEE div expansion.
__device__ __forceinline__ float fast_rcp(float x) {
#if defined(__HIP_DEVICE_COMPILE__) && __has_builtin(__builtin_amdgcn_rcpf)
  return __builtin_amdgcn_rcpf(x);
#else
  return 1.0f / x;
#endif
}

#if HAVE_TDM
// 1-D TDM copy: ndwords DWORDs from global to LDS (D# groups per CDNA5 ISA 8.3/8.4).
// This toolchain exposes the 6-arg builtin: (g0, g1, g2, g3, g4, cpol).
__device__ __forceinline__ void tdm_load_1d(unsigned int lds_byte_off,
                                            const void* gsrc,
                                            unsigned int ndwords) {
  unsigned long long ga = (unsigned long long)(uintptr_t)gsrc;
  u32x4 g0 = {};
  g0[0] = 1u;                                             // count=1 (valid), user mode
  g0[1] = lds_byte_off;                                   // lds_addr
  g0[2] = (unsigned int)(ga & 0xffffffffull);             // global_addr[31:0]
  g0[3] = ((unsigned int)(ga >> 32) & 0x01ffffffu) | (2u << 30); // addr[56:32] | type=2
  i32x8 g1 = {};
  g1[0] = (2 << 16);                                      // data_size=4B, no multicast
  g1[1] = (int)((ndwords & 0xffffu) << 16);               // tensor_dim0[15:0]
  g1[2] = (int)(((ndwords >> 16) & 0xffffu) | (1u << 16));// tensor_dim0[31:16], tensor_dim1=1
  g1[3] = (int)((ndwords & 0xffffu) << 16);               // tile_dim0 = ndwords
  g1[4] = 0;                                              // tile_dim1/2 unused
  g1[5] = (int)ndwords;                                   // tensor_dim0_stride lo
  g1[6] = 0;
  g1[7] = 0;
  i32x4 gz4 = {};
  i32x8 gz8 = {};
  __builtin_amdgcn_tensor_load_to_lds(g0, g1, gz4, gz4, gz8, 0);
}
__device__ __forceinline__ void tdm_wait0() {
#if __has_builtin(__builtin_amdgcn_s_wait_tensorcnt)
  __builtin_amdgcn_s_wait_tensorcnt(0);
#else
  asm volatile("s_wait_tensorcnt 0x0" ::: "memory");
#endif
  asm volatile("" ::: "memory");
}
#endif

// ---------------------------------------------------------------------------
// K0: zero int region (hist + thrbin + cnt)
__global__ void zero_i32(int* __restrict__ p, int n) {
  int i = blockIdx.x * 256 + threadIdx.x;
  if (i < n) p[i] = 0;
}

// ---------------------------------------------------------------------------
// K1: decode anchors+deltas -> normalized boxes (B,N,4)
__global__ void decode_boxes(const float* __restrict__ deltas,
                             const float* __restrict__ anchors,
                             float* __restrict__ boxes, int N) {
  int i = blockIdx.x * 256 + threadIdx.x;
  int total = BATCH * N;
  if (i >= total) return;
  int n = i % N;
  const float* a = anchors + (size_t)n * 4;
  const float* d = deltas + (size_t)i * 4;
  float ay1 = a[0], ax1 = a[1], ay2 = a[2], ax2 = a[3];
  float ah = ay2 - ay1, aw = ax2 - ax1;
  float acy = (ay1 + ay2) * 0.5f, acx = (ax1 + ax2) * 0.5f;
  float cy = d[0] * ah + acy, cx = d[1] * aw + acx;
  float h = __expf(d[2]) * ah, w = __expf(d[3]) * aw;
  const float inv = 1.0f / 512.0f;
  float* o = boxes + (size_t)i * 4;
  o[0] = (cy - h * 0.5f) * inv;
  o[1] = (cx - w * 0.5f) * inv;
  o[2] = (cy + h * 0.5f) * inv;
  o[3] = (cx + w * 0.5f) * inv;
}

// ---------------------------------------------------------------------------
// K2: coalesced per-(batch,class) 128-bin histogram on monotone float keys.
// LDS-privatized 90x128 histogram per workgroup (45KB of the 320KB WGP LDS).
__global__ void class_hist(const float* __restrict__ labels,
                           int* __restrict__ hist, int N, int perBatch) {
  __shared__ unsigned int lh[NCLS * NBINS];
  int tid = threadIdx.x;
  for (int i = tid; i < NCLS * NBINS; i += 256) lh[i] = 0u;
  __syncthreads();
  int b = blockIdx.x / perBatch;
  int chunk = blockIdx.x % perBatch;
  long long rowbase = (long long)b * N * NCLS;
  long long base = rowbase + (long long)chunk * CHUNK;
  long long lim = rowbase + (long long)N * NCLS;
  for (int k = 0; k < 256; ++k) {
    long long e = base + (long long)k * 256 + tid;
    if (e < lim) {
      float v = labels[e];
      int c = (int)((e - rowbase) % NCLS);
      unsigned int bin = monokey(v) >> 25;
      atomicAdd(&lh[c * NBINS + bin], 1u);
    }
  }
  __syncthreads();
  for (int i = tid; i < NCLS * NBINS; i += 256) {
    unsigned int v = lh[i];
    if (v) atomicAdd((unsigned int*)&hist[((size_t)b * NCLS + (i >> 7)) * NBINS + (i & (NBINS - 1))], v);
  }
}

// ---------------------------------------------------------------------------
// K3: per row, largest bin t such that count(bin >= t) >= TOPK
__global__ void pick_threshold(const int* __restrict__ hist, int* __restrict__ thrb) {
  int r = blockIdx.x * 256 + threadIdx.x;
  if (r >= NROWS) return;
  const int* h = hist + (size_t)r * NBINS;
  int cum = 0, t = 0;
  for (int bkt = NBINS - 1; bkt >= 0; --bkt) {
    cum += h[bkt];
    if (cum >= TOPK) { t = bkt; break; }
  }
  thrb[r] = t;
}

// ---------------------------------------------------------------------------
// K4: coalesced compaction of candidates (logit, anchor idx) above the bin thr
__global__ void compact_cand(const float* __restrict__ labels,
                             const int* __restrict__ thrb,
                             int* __restrict__ cnt,
                             float* __restrict__ cand_s,
                             int* __restrict__ cand_i, int N, int perBatch) {
  __shared__ int th_s[NCLS];
  int tid = threadIdx.x;
  int b = blockIdx.x / perBatch;
  if (tid < NCLS) th_s[tid] = thrb[b * NCLS + tid];
  __syncthreads();
  long long rowbase = (long long)b * N * NCLS;
  long long base = rowbase + (long long)(blockIdx.x % perBatch) * CHUNK;
  long long lim = rowbase + (long long)N * NCLS;
  for (int k = 0; k < 256; ++k) {
    long long e = base + (long long)k * 256 + tid;
    if (e < lim) {
      __builtin_prefetch(&labels[e + CHUNK], 0, 0);  // global_prefetch_b8
      float v = labels[e];
      long long rel = e - rowbase;
      int c = (int)(rel % NCLS);
      int n = (int)(rel / NCLS);
      int bin = (int)(monokey(v) >> 25);
      if (bin >= th_s[c]) {
        int row = b * NCLS + c;
        int pos = atomicAdd(&cnt[row], 1);
        if (pos < CAP) {
          cand_s[(size_t)row * CAP + pos] = v;
          cand_i[(size_t)row * CAP + pos] = n;
        }
      }
    }
  }
}

// ---------------------------------------------------------------------------
// K5: per-(batch,class) top-200 + fast-NMS. TDM stages candidates to LDS;
// 16x16 IoU tiles use V_WMMA_F32_16X16X4_F32 for the pairwise union base
// (area_i + area_j == [area_i,1] x [1;area_j], a K=2 matmul in the K=4 slot).
__global__ void topk_nms(const float* __restrict__ boxes,
                         const float* __restrict__ cand_s,
                         const int* __restrict__ cand_i,
                         const int* __restrict__ cnt,
                         float* __restrict__ nms_s,
                         float* __restrict__ nms_b, int N) {
  __shared__ float s_cs[CAP];
  __shared__ int   s_ci[CAP];
  __shared__ float s_sc[TOPKP];
  __shared__ int   s_id[TOPKP];
  __shared__ float s_box[TOPKP][4];
  __shared__ float s_area[TOPKP];
  __shared__ float rs[256];
  __shared__ int   ri[256];

  int row = blockIdx.x;
  int b = row / NCLS;
  int tid = threadIdx.x;
  int m = cnt[row];
  if (m > CAP) m = CAP;

#if HAVE_TDM
  if (tid == 0) {
    tdm_load_1d((unsigned int)(uintptr_t)(void*)s_cs, cand_s + (size_t)row * CAP, CAP);
    tdm_load_1d((unsigned int)(uintptr_t)(void*)s_ci, cand_i + (size_t)row * CAP, CAP);
    tdm_wait0();
  }
#else
  for (int i = tid; i < CAP; i += 256) {
    s_cs[i] = cand_s[(size_t)row * CAP + i];
    s_ci[i] = cand_i[(size_t)row * CAP + i];
  }
#endif
  __syncthreads();

  // sequential top-200 by workgroup argmax over LDS
  for (int t = 0; t < TOPK; ++t) {
    float bv = -3.0e38f;
    int bi = -1;
    for (int i = tid; i < m; i += 256) {
      float v = s_cs[i];
      if (v > bv) { bv = v; bi = i; }
    }
    rs[tid] = bv; ri[tid] = bi;
    __syncthreads();
    for (int s = 128; s > 0; s >>= 1) {
      if (tid < s) {
        if (rs[tid + s] > rs[tid]) { rs[tid] = rs[tid + s]; ri[tid] = ri[tid + s]; }
      }
      __syncthreads();
    }
    if (tid == 0) {
      int w = ri[0];
      if (w >= 0) { s_sc[t] = rs[0]; s_id[t] = s_ci[w]; s_cs[w] = -3.0e38f; }
      else        { s_sc[t] = -3.0e38f; s_id[t] = -1; }
    }
    __syncthreads();
  }

  // gather candidate boxes + areas (pad 200..207 with degenerate boxes)
  if (tid < TOPKP) {
    int a = (tid < TOPK) ? s_id[tid] : -1;
    float x0 = 0.f, x1 = 0.f, x2 = 0.f, x3 = 0.f;
    if (a >= 0) {
      const float* bp = boxes + ((size_t)b * N + a) * 4;
      x0 = bp[0]; x1 = bp[1]; x2 = bp[2]; x3 = bp[3];
    }
    if (tid >= TOPK) { s_sc[tid] = -3.0e38f; s_id[tid] = -1; }
    s_box[tid][0] = x0; s_box[tid][1] = x1; s_box[tid][2] = x2; s_box[tid][3] = x3;
    s_area[tid] = (x2 - x0) * (x3 - x1);
  }
  __syncthreads();

  // tiled triangular NMS: one wave per 16-column tile J, rows I <= J.
  // Branchless inner body: unconditional ds_load_b128 + VALU, single v_rcp.
  int w = tid >> 5;
  int lane = tid & 31;
  for (int J = w; J < NTILE; J += 8) {
    float colmax = 0.f;
    int nc = lane & 15;
    int j = J * 16 + nc;
    float bj0 = s_box[j][0], bj1 = s_box[j][1], bj2 = s_box[j][2], bj3 = s_box[j][3];
    v2f bop;
    bop.x = (lane < 16) ? 1.0f : s_area[J * 16 + (lane - 16)];
    bop.y = 0.f;
    for (int I = 0; I <= J; ++I) {
      float ub[8];
#if HAVE_WMMA4
      v2f aop;
      aop.x = (lane < 16) ? s_area[I * 16 + lane] : 0.f;
      aop.y = (lane < 16) ? 1.f : 0.f;
      v8f cz = {0.f, 0.f, 0.f, 0.f, 0.f, 0.f, 0.f, 0.f};
      v8f d = __builtin_amdgcn_wmma_f32_16x16x4_f32(
          false, aop, false, bop, (short)0, cz, false, false);
#pragma unroll
      for (int r = 0; r < 8; ++r) ub[r] = d[r];
#else
#pragma unroll
      for (int r = 0; r < 8; ++r)
        ub[r] = s_area[I * 16 + r + ((lane < 16) ? 0 : 8)] + s_area[j];
#endif
#pragma unroll
      for (int r = 0; r < 8; ++r) {
        int mrow = r + ((lane < 16) ? 0 : 8);
        int i = I * 16 + mrow;
        float iy1 = fmaxf(s_box[i][0], bj0);
        float ix1 = fmaxf(s_box[i][1], bj1);
        float iy2 = fminf(s_box[i][2], bj2);
        float ix2 = fminf(s_box[i][3], bj3);
        float inter = fmaxf(iy2 - iy1, 0.f) * fmaxf(ix2 - ix1, 0.f);
        float uni = fmaxf(ub[r] - inter, 1e-8f);
        float iou = inter * fast_rcp(uni);
        iou = (i < j) ? iou : 0.f;          // branchless triangular mask
        colmax = fmaxf(colmax, iou);
      }
    }
    colmax = fmaxf(colmax, __shfl_xor(colmax, 16, 32));
    if (lane < 16) {
      int jj = J * 16 + lane;
      if (jj < TOPK) {
        float s = s_sc[jj];
        float outp = 0.f;
        if (colmax <= 0.5f && s > LOGIT_THR)
          outp = 1.f / (1.f + __expf(-s));
        size_t o = (size_t)row * TOPK + jj;
        nms_s[o] = outp;
        nms_b[o * 4 + 0] = s_box[jj][0];
        nms_b[o * 4 + 1] = s_box[jj][1];
        nms_b[o * 4 + 2] = s_box[jj][2];
        nms_b[o * 4 + 3] = s_box[jj][3];
      }
    }
  }
}

// ---------------------------------------------------------------------------
// K6: per batch, top-100 of the 18000 NMS scores -> final outputs.
// out layout: [0,6400) boxes, [6400,8000) scores, [8000,9600) classes (float)
__global__ void final_topk(float* __restrict__ nms_s,
                           const float* __restrict__ nms_b,
                           float* __restrict__ out) {
  __shared__ float rs[256];
  __shared__ int   ri[256];
  int b = blockIdx.x;
  int tid = threadIdx.x;
  float* sc = nms_s + (size_t)b * NCLS * TOPK;
  const int NFLAT = NCLS * TOPK;
  for (int t = 0; t < MAXDET; ++t) {
    float bv = -3.0e38f;
    int bi = 0;
    for (int i = tid; i < NFLAT; i += 256) {
      float v = sc[i];
      if (v > bv) { bv = v; bi = i; }
    }
    rs[tid] = bv; ri[tid] = bi;
    __syncthreads();
    for (int s = 128; s > 0; s >>= 1) {
      if (tid < s) {
        if (rs[tid + s] > rs[tid]) { rs[tid] = rs[tid + s]; ri[tid] = ri[tid + s]; }
      }
      __syncthreads();
    }
    if (tid == 0) {
      int f = ri[0];
      float v = rs[0];
      if (v < 0.f) v = 0.f;
      const float* bb = nms_b + ((size_t)b * NCLS * TOPK + f) * 4;
      float* ob = out + ((size_t)b * MAXDET + t) * 4;
      ob[0] = bb[0]; ob[1] = bb[1]; ob[2] = bb[2]; ob[3] = bb[3];
      out[BATCH * MAXDET * 4 + b * MAXDET + t] = v;
      out[BATCH * MAXDET * 5 + b * MAXDET + t] = (float)(f / TOPK);
      sc[f] = -1.f;  // rewritten fresh by topk_nms each launch
    }
    __syncthreads();
  }
}

// ---------------------------------------------------------------------------
extern "C" void kernel_launch(void* const* d_in, const int* in_sizes, int n_in,
                              void* d_out, int out_size, void* d_ws, size_t ws_size,
                              hipStream_t stream) {
  const float* d_deltas = (const float*)d_in[0];   // (B,N,4)
  const float* d_labels = (const float*)d_in[1];   // (B,N,90)
  const float* d_anch   = (const float*)d_in[2];   // (N,4)
  int N = in_sizes[2] / 4;

  // workspace layout (all 4-byte units)
  float* w_boxes = (float*)d_ws;                               // B*N*4
  float* w_nms_s = w_boxes + (size_t)BATCH * N * 4;            // NROWS*200
  float* w_nms_b = w_nms_s + (size_t)NROWS * TOPK;             // NROWS*200*4
  float* w_cand_s = w_nms_b + (size_t)NROWS * TOPK * 4;        // NROWS*CAP
  int*   w_cand_i = (int*)(w_cand_s + (size_t)NROWS * CAP);    // NROWS*CAP
  int*   w_hist  = w_cand_i + (size_t)NROWS * CAP;             // NROWS*NBINS
  int*   w_thr   = w_hist + (size_t)NROWS * NBINS;             // NROWS
  int*   w_cnt   = w_thr + NROWS;                              // NROWS

  int zeroN = NROWS * NBINS + 2 * NROWS;
  hipLaunchKernelGGL(zero_i32, dim3((zeroN + 255) / 256), dim3(256), 0, stream,
                     w_hist, zeroN);

  int tot = BATCH * N;
  hipLaunchKernelGGL(decode_boxes, dim3((tot + 255) / 256), dim3(256), 0, stream,
                     d_deltas, d_anch, w_boxes, N);

  int perBatch = (N * NCLS + CHUNK - 1) / CHUNK;
  hipLaunchKernelGGL(class_hist, dim3(BATCH * perBatch), dim3(256), 0, stream,
                     d_labels, w_hist, N, perBatch);

  hipLaunchKernelGGL(pick_threshold, dim3((NROWS + 255) / 256), dim3(256), 0, stream,
                     w_hist, w_thr);

  hipLaunchKernelGGL(compact_cand, dim3(BATCH * perBatch), dim3(256), 0, stream,
                     d_labels, w_thr, w_cnt, w_cand_s, w_cand_i, N, perBatch);

  hipLaunchKernelGGL(topk_nms, dim3(NROWS), dim3(256), 0, stream,
                     w_boxes, w_cand_s, w_cand_i, w_cnt, w_nms_s, w_nms_b, N);

  hipLaunchKernelGGL(final_topk, dim3(BATCH), dim3(256), 0, stream,
                     w_nms_s, w_nms_b, (float*)d_out);
}